// CoAttentionModule_73289321939610
// MI455X (gfx1250) — compile-verified
//
#include <hip/hip_runtime.h>

typedef __attribute__((ext_vector_type(2))) float v2f;
typedef __attribute__((ext_vector_type(8))) float v8f;

#define B_   4
#define N1_  256
#define N2_  256
#define D_   512
#define DC   256           // D-chunk held in LDS per pass
#define LOG2E 1.44269504088896340736f

// CDNA5 has a hardware V_TANH_F32 (listed as a transcendental VALU op in the
// ISA). Use it if clang exposes the builtin; otherwise fall back to the
// exp2/rcp formulation (2 trans + 3 VALU instead of 1 trans).
#if __has_builtin(__builtin_amdgcn_tanhf)
__device__ __forceinline__ float tanh_fast(float x) {
  return __builtin_amdgcn_tanhf(x);
}
#else
__device__ __forceinline__ float tanh_fast(float x) {
  float e = __builtin_amdgcn_exp2f(x * (2.0f * LOG2E));
  float r = __builtin_amdgcn_rcpf(e + 1.0f);
  return __builtin_fmaf(-2.0f, r, 1.0f);
}
#endif

// ---------------------------------------------------------------------------
// Stage 1: A[b,i,j] = sum_d tanh(m1[b,i,d] + m2[b,j,d]) * w[d] + bias
// One 256-thread block per 16x16 (i,j) tile; LDS-tiled over D in chunks of 256.
// Row pad of 4 floats => s2[tj][d] readers hit 16 distinct banks (4-bank skew).
// ---------------------------------------------------------------------------
__global__ void scores_kernel(const float* __restrict__ m1,
                              const float* __restrict__ m2,
                              const float* __restrict__ w,
                              const float* __restrict__ bias,
                              float* __restrict__ A) {
  __shared__ __align__(16) float s1[16][DC + 4];
  __shared__ __align__(16) float s2[16][DC + 4];
  __shared__ __align__(16) float sw[DC];
  const int tid = threadIdx.x;
  const int ti = tid >> 4, tj = tid & 15;
  const int i0 = blockIdx.x * 16, j0 = blockIdx.y * 16, b = blockIdx.z;

  float acc = 0.0f;
  for (int ch = 0; ch < D_ / DC; ++ch) {
    sw[tid] = w[ch * DC + tid];
    // 16 rows x 64 float4 per matrix, 256 threads -> 4 float4 each, coalesced.
    for (int idx = tid; idx < 16 * (DC / 4); idx += 256) {
      int row = idx >> 6, c4 = (idx & 63) * 4;
      *(float4*)&s1[row][c4] =
          *(const float4*)&m1[((size_t)(b * N1_ + i0 + row)) * D_ + ch * DC + c4];
      *(float4*)&s2[row][c4] =
          *(const float4*)&m2[((size_t)(b * N2_ + j0 + row)) * D_ + ch * DC + c4];
    }
    __syncthreads();
#pragma unroll 4
    for (int d = 0; d < DC; d += 4) {
      float4 x1 = *(const float4*)&s1[ti][d];
      float4 x2 = *(const float4*)&s2[tj][d];
      float4 wv = *(const float4*)&sw[d];
      acc = __builtin_fmaf(tanh_fast(x1.x + x2.x), wv.x, acc);
      acc = __builtin_fmaf(tanh_fast(x1.y + x2.y), wv.y, acc);
      acc = __builtin_fmaf(tanh_fast(x1.z + x2.z), wv.z, acc);
      acc = __builtin_fmaf(tanh_fast(x1.w + x2.w), wv.w, acc);
    }
    __syncthreads();
  }
  A[((size_t)(b * N1_ + i0 + ti)) * N2_ + j0 + tj] = acc + bias[0];
}

// ---------------------------------------------------------------------------
// Stage 2a: softmax over j (axis=2). One 256-thread block per row.
// ---------------------------------------------------------------------------
__global__ void row_softmax_kernel(const float* __restrict__ A,
                                   float* __restrict__ P) {
  __shared__ float red[256];
  const int t = threadIdx.x;
  const size_t row = blockIdx.x;  // b*N1 + i
  float v = A[row * N2_ + t];
  red[t] = v; __syncthreads();
  for (int s = 128; s > 0; s >>= 1) {
    if (t < s) red[t] = fmaxf(red[t], red[t + s]);
    __syncthreads();
  }
  float mx = red[0]; __syncthreads();
  float e = __builtin_amdgcn_exp2f((v - mx) * LOG2E);
  red[t] = e; __syncthreads();
  for (int s = 128; s > 0; s >>= 1) {
    if (t < s) red[t] += red[t + s];
    __syncthreads();
  }
  P[row * N2_ + t] = e * (1.0f / red[0]);
}

// ---------------------------------------------------------------------------
// Stage 2b: softmax over i (axis=1). Thread = column j, online max/sum over i;
// all loads coalesced across j.
// ---------------------------------------------------------------------------
__global__ void col_softmax_kernel(const float* __restrict__ A,
                                   float* __restrict__ P) {
  const int b = blockIdx.x;
  const int j = threadIdx.x;
  const float* Ab = A + (size_t)b * N1_ * N2_;
  float* Pb       = P + (size_t)b * N1_ * N2_;
  float m = -INFINITY, s = 0.0f;
  for (int i = 0; i < N1_; ++i) {
    float v = Ab[(size_t)i * N2_ + j];
    float nm = fmaxf(m, v);
    s = s * __builtin_amdgcn_exp2f((m - nm) * LOG2E) +
        __builtin_amdgcn_exp2f((v - nm) * LOG2E);
    m = nm;
  }
  float inv = 1.0f / s;
  for (int i = 0; i < N1_; ++i) {
    float v = Ab[(size_t)i * N2_ + j];
    Pb[(size_t)i * N2_ + j] = __builtin_amdgcn_exp2f((v - m) * LOG2E) * inv;
  }
}

// ---------------------------------------------------------------------------
// Stage 3: C[b] = op(A[b]) (256x256) @ B[b] (256x512) with V_WMMA_F32_16X16X4.
// One wave per 16x32 output tile (A-frag reused across two B-frags).
// A layout: lane -> M = lane%16, K-pair = 2*(lane/16). B layout: lane -> N =
// lane%16, K-pair = 2*(lane/16). C/D: VGPR r -> row r + 8*(lane/16), col lane%16.
// ---------------------------------------------------------------------------
template <bool TRANSA>
__global__ void gemm_wmma_kernel(const float* __restrict__ Amat,  // [256,256]
                                 const float* __restrict__ Bmat,  // [256,512]
                                 float* __restrict__ Cmat) {      // [256,512]
  const int lane = threadIdx.x & 31;
  const int mN   = lane & 15;
  const int half = lane >> 4;
  const int k0   = half * 2;
  const int i0   = blockIdx.x * 16;
  const int n0   = blockIdx.y * 32;
  const int b    = blockIdx.z;
  const float* Ab = Amat + (size_t)b * 256 * 256;
  const float* Bb = Bmat + (size_t)b * 256 * 512;
  float*       Cb = Cmat + (size_t)b * 256 * 512;

  v8f acc0 = {0.f, 0.f, 0.f, 0.f, 0.f, 0.f, 0.f, 0.f};
  v8f acc1 = acc0;
  for (int k = 0; k < 256; k += 4) {
    v2f a;
    if (TRANSA) {  // A[m][kk] = Amat[kk][i0+m]
      a.x = Ab[(size_t)(k + k0)     * 256 + (i0 + mN)];
      a.y = Ab[(size_t)(k + k0 + 1) * 256 + (i0 + mN)];
    } else {       // A[m][kk] = Amat[i0+m][kk]
      a.x = Ab[(size_t)(i0 + mN) * 256 + (k + k0)];
      a.y = Ab[(size_t)(i0 + mN) * 256 + (k + k0 + 1)];
    }
    v2f b0, b1;
    b0.x = Bb[(size_t)(k + k0)     * 512 + n0 + mN];
    b0.y = Bb[(size_t)(k + k0 + 1) * 512 + n0 + mN];
    b1.x = Bb[(size_t)(k + k0)     * 512 + n0 + 16 + mN];
    b1.y = Bb[(size_t)(k + k0 + 1) * 512 + n0 + 16 + mN];
    acc0 = __builtin_amdgcn_wmma_f32_16x16x4_f32(false, a, false, b0,
                                                 (short)0, acc0, false, false);
    acc1 = __builtin_amdgcn_wmma_f32_16x16x4_f32(false, a, false, b1,
                                                 (short)0, acc1, false, false);
  }
#pragma unroll
  for (int r = 0; r < 8; ++r) {
    int row = i0 + r + half * 8;
    Cb[(size_t)row * 512 + n0 + mN]      = acc0[r];
    Cb[(size_t)row * 512 + n0 + 16 + mN] = acc1[r];
  }
}

// ---------------------------------------------------------------------------
extern "C" void kernel_launch(void* const* d_in, const int* in_sizes, int n_in,
                              void* d_out, int out_size, void* d_ws, size_t ws_size,
                              hipStream_t stream) {
  const float* modal1 = (const float*)d_in[0];  // [4,256,512]
  const float* modal2 = (const float*)d_in[1];  // [4,256,512]
  const float* w      = (const float*)d_in[2];  // [512]
  const float* bias   = (const float*)d_in[3];  // scalar

  float* A  = (float*)d_ws;                       // [4,256,256] logits
  float* P1 = A  + (size_t)B_ * N1_ * N2_;        // softmax over j
  float* P2 = P1 + (size_t)B_ * N1_ * N2_;        // softmax over i

  float* out1 = (float*)d_out;                    // co_feature1 [4,256,512]
  float* out2 = out1 + (size_t)B_ * N1_ * D_;     // co_feature2 [4,256,512]

  scores_kernel<<<dim3(N1_ / 16, N2_ / 16, B_), 256, 0, stream>>>(
      modal1, modal2, w, bias, A);
  row_softmax_kernel<<<dim3(B_ * N1_), 256, 0, stream>>>(A, P1);
  col_softmax_kernel<<<dim3(B_), 256, 0, stream>>>(A, P2);
  // co_feature1 = P1 @ modal2
  gemm_wmma_kernel<false><<<dim3(16, 16, B_), 32, 0, stream>>>(P1, modal2, out1);
  // co_feature2 = P2^T @ modal1
  gemm_wmma_kernel<true ><<<dim3(16, 16, B_), 32, 0, stream>>>(P2, modal1, out2);
}